// DBFusion_77936476553546
// MI455X (gfx1250) — compile-verified
//
#include <hip/hip_runtime.h>
#include <hip/hip_bf16.h>

typedef __attribute__((ext_vector_type(2))) float v2f;
typedef __attribute__((ext_vector_type(8))) float v8f;

#define CCH   64
#define LLEN  65536
#define TILE  128
#define WS    65      // LDS stride for 64x64 weight tiles (padded)
#define TS    132     // LDS stride for 64x128 data tiles (padded, 16B-aligned rows)
#define NPOS  4194304.0f  // C*L elements per batch (2^22)

__global__ __launch_bounds__(256) void zero_stats_k(float* stats) {
  if (threadIdx.x < 16) stats[threadIdx.x] = 0.0f;
}

__global__ __launch_bounds__(256) void fused_gemm_pass(
    const float* __restrict__ x, const float* __restrict__ inj0,
    const float* __restrict__ inj1, const float* __restrict__ resid,
    const float* __restrict__ gate_w, const float* __restrict__ gate_b,
    const float* __restrict__ fuse_w, const float* __restrict__ fuse_b,
    float* __restrict__ out_pre, float* __restrict__ stats)
{
  __shared__ __align__(16) float sWg[CCH * WS];
  __shared__ __align__(16) float sWf[CCH * WS];
  __shared__ __align__(16) float sS [CCH * TS];
  __shared__ __align__(16) float sX [CCH * TS];
  __shared__ float sBg[CCH];
  __shared__ float sBf[CCH];

  const int tid = threadIdx.x;
  const int b   = blockIdx.x >> 9;              // 512 tiles per batch
  const int l0  = (blockIdx.x & 511) * TILE;
  const size_t base = ((size_t)b * CCH) * (size_t)LLEN + (size_t)l0;

  // ---- stage weights (row-major, padded stride)
  #pragma unroll
  for (int i = 0; i < 16; ++i) {
    const int idx = tid + i * 256;
    const int r = idx >> 6, cc = idx & 63;
    sWg[r * WS + cc] = gate_w[idx];
    sWf[r * WS + cc] = fuse_w[idx];
  }
  if (tid < CCH) { sBg[tid] = 2.0f * gate_b[tid]; sBf[tid] = fuse_b[tid]; }

  // ---- stage x and s = inj0 + inj1 + 2*x (64 ch x 128 pos), coalesced float4
  #pragma unroll
  for (int i = 0; i < 8; ++i) {
    const int idx4 = tid + i * 256;
    const int r = idx4 >> 5, c4 = (idx4 & 31) << 2;
    const size_t g = base + (size_t)r * LLEN + (size_t)c4;
    const float4 vx = *(const float4*)(x    + g);
    const float4 v0 = *(const float4*)(inj0 + g);
    const float4 v1 = *(const float4*)(inj1 + g);
    float4 vs;
    vs.x = v0.x + v1.x + 2.0f * vx.x;
    vs.y = v0.y + v1.y + 2.0f * vx.y;
    vs.z = v0.z + v1.z + 2.0f * vx.z;
    vs.w = v0.w + v1.w + 2.0f * vx.w;
    *(float4*)&sX[r * TS + c4] = vx;
    *(float4*)&sS[r * TS + c4] = vs;
  }
  __syncthreads();

  const int lane = tid & 31;
  const int lo   = lane & 15;
  const int hi   = lane >> 4;     // half-wave select
  const int wv   = tid >> 5;
  const int col  = wv * 16 + lo;  // this wave's column for B / C / D frags
  const int kb   = hi * 2;        // K base within a 4-wide K step (A/B layout)

  // ---- GEMM1: g = gate_w @ s   (64x64 @ 64x16 per wave, WMMA f32 16x16x4)
  v8f acc[4];
  #pragma unroll
  for (int mt = 0; mt < 4; ++mt) acc[mt] = (v8f)0.0f;

  #pragma unroll
  for (int ks = 0; ks < 16; ++ks) {
    const int c0 = ks * 4 + kb;
    v2f bf; bf.x = sS[c0 * TS + col]; bf.y = sS[(c0 + 1) * TS + col];
    #pragma unroll
    for (int mt = 0; mt < 4; ++mt) {
      const int r = mt * 16 + lo;
      v2f af; af.x = sWg[r * WS + c0]; af.y = sWg[r * WS + c0 + 1];
      acc[mt] = __builtin_amdgcn_wmma_f32_16x16x4_f32(
          false, af, false, bf, (short)0, acc[mt], false, false);
    }
  }

  // a = x * (g + 2*gate_b); write in place of s (own columns only)
  #pragma unroll
  for (int mt = 0; mt < 4; ++mt) {
    #pragma unroll
    for (int j = 0; j < 8; ++j) {
      const int ch = mt * 16 + j + 8 * hi;   // C/D layout: VGPR j -> M=j / j+8
      const float gch = acc[mt][j] + sBg[ch];
      sS[ch * TS + col] = sX[ch * TS + col] * gch;
    }
  }
  __syncthreads();

  // ---- stage residual over sX (x no longer needed)
  #pragma unroll
  for (int i = 0; i < 8; ++i) {
    const int idx4 = tid + i * 256;
    const int r = idx4 >> 5, c4 = (idx4 & 31) << 2;
    const size_t g = base + (size_t)r * LLEN + (size_t)c4;
    *(float4*)&sX[r * TS + c4] = *(const float4*)(resid + g);
  }
  __syncthreads();

  // ---- GEMM2: p = fuse_w @ a
  #pragma unroll
  for (int mt = 0; mt < 4; ++mt) acc[mt] = (v8f)0.0f;
  #pragma unroll
  for (int ks = 0; ks < 16; ++ks) {
    const int c0 = ks * 4 + kb;
    v2f bf; bf.x = sS[c0 * TS + col]; bf.y = sS[(c0 + 1) * TS + col];
    #pragma unroll
    for (int mt = 0; mt < 4; ++mt) {
      const int r = mt * 16 + lo;
      v2f af; af.x = sWf[r * WS + c0]; af.y = sWf[r * WS + c0 + 1];
      acc[mt] = __builtin_amdgcn_wmma_f32_16x16x4_f32(
          false, af, false, bf, (short)0, acc[mt], false, false);
    }
  }

  // o = relu(p + fuse_b + residual); write in place (own columns)
  #pragma unroll
  for (int mt = 0; mt < 4; ++mt) {
    #pragma unroll
    for (int j = 0; j < 8; ++j) {
      const int ch = mt * 16 + j + 8 * hi;
      const float p = acc[mt][j] + sBf[ch] + sX[ch * TS + col];
      sS[ch * TS + col] = fmaxf(p, 0.0f);
    }
  }
  __syncthreads();

  // ---- coalesced float4 store of pre-GN result + per-batch stats
  float ps = 0.0f, pq = 0.0f;
  #pragma unroll
  for (int i = 0; i < 8; ++i) {
    const int idx4 = tid + i * 256;
    const int r = idx4 >> 5, c4 = (idx4 & 31) << 2;
    const float4 v = *(const float4*)&sS[r * TS + c4];
    *(float4*)(out_pre + base + (size_t)r * LLEN + (size_t)c4) = v;
    ps += v.x + v.y + v.z + v.w;
    pq += v.x * v.x + v.y * v.y + v.z * v.z + v.w * v.w;
  }
  #pragma unroll
  for (int off = 16; off > 0; off >>= 1) {
    ps += __shfl_xor(ps, off, 32);
    pq += __shfl_xor(pq, off, 32);
  }
  if (lane == 0) {
    atomicAdd(&stats[2 * b + 0], ps);
    atomicAdd(&stats[2 * b + 1], pq);
  }
}

__global__ __launch_bounds__(256) void groupnorm_pass(
    float* __restrict__ out, const float* __restrict__ stats,
    const float* __restrict__ gn_w, const float* __restrict__ gn_b)
{
  const size_t i4 = (size_t)blockIdx.x * 256 + threadIdx.x;
  const size_t e  = i4 << 2;                  // element offset (float4 granular)
  const int b  = (int)(e >> 22);              // 2^22 elements per batch
  const int ch = (int)((e >> 16) & 63);       // 2^16 positions per channel
  const float inv_n = 1.0f / NPOS;
  const float mean = stats[2 * b] * inv_n;
  const float var  = stats[2 * b + 1] * inv_n - mean * mean;
  const float rstd = rsqrtf(var + 1e-5f);
  const float sc = rstd * gn_w[ch];
  const float sh = gn_b[ch] - mean * sc;
  float4 v = *(float4*)(out + e);
  v.x = v.x * sc + sh;
  v.y = v.y * sc + sh;
  v.z = v.z * sc + sh;
  v.w = v.w * sc + sh;
  *(float4*)(out + e) = v;
}

extern "C" void kernel_launch(void* const* d_in, const int* in_sizes, int n_in,
                              void* d_out, int out_size, void* d_ws, size_t ws_size,
                              hipStream_t stream) {
  (void)in_sizes; (void)n_in; (void)out_size; (void)ws_size;
  const float* x      = (const float*)d_in[0];
  const float* inj0   = (const float*)d_in[1];
  const float* inj1   = (const float*)d_in[2];
  const float* resid  = (const float*)d_in[3];
  const float* gate_w = (const float*)d_in[4];
  const float* gate_b = (const float*)d_in[5];
  const float* fuse_w = (const float*)d_in[6];
  const float* fuse_b = (const float*)d_in[7];
  const float* gn_w   = (const float*)d_in[8];
  const float* gn_b   = (const float*)d_in[9];
  float* out   = (float*)d_out;
  float* stats = (float*)d_ws;   // 16 floats: {sum, sumsq} x 8 batches

  zero_stats_k<<<1, 32, 0, stream>>>(stats);
  fused_gemm_pass<<<4096, 256, 0, stream>>>(
      x, inj0, inj1, resid, gate_w, gate_b, fuse_w, fuse_b, out, stats);
  groupnorm_pass<<<32768, 256, 0, stream>>>(out, stats, gn_w, gn_b);
}